// GraphConnection_80650895884490
// MI455X (gfx1250) — compile-verified
//
#include <hip/hip_runtime.h>

// ---------------------------------------------------------------------------
// MI455X / gfx1250. All dense linear algebra runs through
// V_WMMA_F32_16X16X4_F32 (wave32). Each wave owns a 16x32 C tile (two v8f
// accumulators sharing one A fragment), K unrolled x4 (8 WMMA / iteration).
// Out-of-range rows/cols are handled by index clamping (their products only
// land in C rows/cols that are never stored), so the hot loop has NO
// divergent branches -- unconditional b64/b32 loads feeding back-to-back
// WMMAs. Problem is bandwidth-bound (~10 GFLOP vs ~600 MB => AI ~16 F/B at
// 23.3 TB/s), so fp32 WMMA keeps reference numerics at no throughput cost.
// ---------------------------------------------------------------------------

typedef __attribute__((ext_vector_type(2))) float v2f;
typedef __attribute__((ext_vector_type(8))) float v8f;

#define LRELU_ALPHA 0.4f
#define BIGMOD (1 << 30)
#define NEG_INF (-3.402823466e38f)

static inline int cdiv(int a, int b) { return (a + b - 1) / b; }

#define WMMA_F32(av, bv, acc) \
  __builtin_amdgcn_wmma_f32_16x16x4_f32(false, (av), false, (bv), (short)0, (acc), false, false)

// ---- batch-index helper: offset = ((z / div) % mod) * stride ---------------
struct BIdx {
  int div; int mod; long stride;
  __device__ __forceinline__ long off(int b) const {
    return (long)((b / div) % mod) * stride;
  }
};

// All pair() loads assume even element offsets (every ld/stride/base offset in
// this file is even), so the contiguous ones lower to global_load_b64.

// ---- A-operand functors: pair(b,m,k) -> (A[m][k], A[m][k+1]) ---------------
struct PlainA {                      // row-major A[m][k]
  const float* __restrict__ p; long ld; BIdx bi;
  __device__ __forceinline__ v2f pair(int b, int m, int k) const {
    return *(const v2f*)(p + bi.off(b) + (long)m * ld + k);
  }
  __device__ __forceinline__ float one(int b, int m, int k) const {
    return p[bi.off(b) + (long)m * ld + k];
  }
};
struct ExpRowA {                     // A[m][k] = exp(raw[b][m][k]-mx[b,m])*inv[b,m]
  const float* __restrict__ p; long ld; long bstride;
  const float* __restrict__ mx; const float* __restrict__ inv; int M;
  __device__ __forceinline__ v2f pair(int b, int m, int k) const {
    int r = b * M + m;
    float mu = mx[r], iv = inv[r];
    v2f raw = *(const v2f*)(p + (long)b * bstride + (long)m * ld + k);
    v2f out; out.x = expf(raw.x - mu) * iv; out.y = expf(raw.y - mu) * iv;
    return out;
  }
  __device__ __forceinline__ float one(int b, int m, int k) const {
    int r = b * M + m;
    return expf(p[(long)b * bstride + (long)m * ld + k] - mx[r]) * inv[r];
  }
};
struct ExpColA {                     // A[m][k] = exp(raw[b][k][m]-mx[b,m])*inv[b,m]
  const float* __restrict__ p; long ld; long bstride;
  const float* __restrict__ mx; const float* __restrict__ inv; int M;
  __device__ __forceinline__ v2f pair(int b, int m, int k) const {
    int r = b * M + m;
    float mu = mx[r], iv = inv[r];
    const float* q = p + (long)b * bstride + (long)k * ld + m;
    v2f out; out.x = expf(q[0] - mu) * iv; out.y = expf(q[ld] - mu) * iv;
    return out;
  }
  __device__ __forceinline__ float one(int b, int m, int k) const {
    int r = b * M + m;
    return expf(p[(long)b * bstride + (long)k * ld + m] - mx[r]) * inv[r];
  }
};
struct SegA {  // virtual concat: stuff (b,Cd,HW) transposed | nstuff (b,HW,So)
  const float* __restrict__ stuff; const float* __restrict__ nstuff;
  int HW; int Cd; int So;
  __device__ __forceinline__ v2f pair(int b, int m, int k) const {
    // pairs are even-aligned and Cd % 4 == 0 => never straddles the boundary
    if (k < Cd) {
      const float* q = stuff + ((long)b * Cd + k) * HW + m;
      v2f out; out.x = q[0]; out.y = q[HW];
      return out;
    }
    return *(const v2f*)(nstuff + ((long)b * HW + m) * So + (k - Cd));
  }
  __device__ __forceinline__ float one(int b, int m, int k) const {
    if (k < Cd) return stuff[((long)b * Cd + k) * HW + m];
    return nstuff[((long)b * HW + m) * So + (k - Cd)];
  }
};

// ---- B-operand functors: pair(b,k,n) -> (B[k][n], B[k+1][n]) ---------------
struct PlainB {                      // row-major B[k][n]
  const float* __restrict__ p; long ld; BIdx bi;
  __device__ __forceinline__ v2f pair(int b, int k, int n) const {
    const float* q = p + bi.off(b) + (long)k * ld + n;
    v2f out; out.x = q[0]; out.y = q[ld];
    return out;
  }
  __device__ __forceinline__ float one(int b, int k, int n) const {
    return p[bi.off(b) + (long)k * ld + n];
  }
};
struct TransB {                      // B[k][n] = raw[b][n][k] (contiguous in k)
  const float* __restrict__ p; long ld; BIdx bi;
  __device__ __forceinline__ v2f pair(int b, int k, int n) const {
    return *(const v2f*)(p + bi.off(b) + (long)n * ld + k);
  }
  __device__ __forceinline__ float one(int b, int k, int n) const {
    return p[bi.off(b) + (long)n * ld + k];
  }
};

// ---- C epilogue: optional bias, ACT 0=none 1=elu ---------------------------
template <int ACT>
struct StoreC {
  float* __restrict__ p; long ld; BIdx bi; const float* __restrict__ bias;
  __device__ __forceinline__ void operator()(int b, int m, int n, float v) const {
    if (bias) v += bias[n];
    if (ACT == 1) v = (v > 0.f) ? v : expm1f(v);
    p[bi.off(b) + (long)m * ld + n] = v;
  }
};

// ---------------------------------------------------------------------------
// WMMA f32 GEMM: block = 8 waves, each wave a 16x32 tile (two 16x16 WMMA
// accumulators sharing the A fragment). Block span 128 rows x 32 cols.
// grid = (ceil(N/32), ceil(M/128), batches).
// Fragment layouts per ISA: A/B k = vgpr + 2*(lane>>4); C row = r + 8*(lane>>4).
// ---------------------------------------------------------------------------
template <typename AF, typename BF, typename CF>
__global__ __launch_bounds__(256) void wmma_gemm(AF A, BF B, CF C,
                                                 int M, int N, int K) {
  const int lane = threadIdx.x & 31;
  const int wave = threadIdx.x >> 5;
  const int b = blockIdx.z;
  const int m0 = blockIdx.y * 128 + wave * 16;
  const int n0 = blockIdx.x * 32;
  if (m0 >= M) return;                 // wave-uniform exit, EXEC stays full
  const int r16 = lane & 15;
  const int grp = lane >> 4;
  // Clamped load indices: out-of-range lanes read row/col 0..M-1/N-1 instead;
  // their products only reach C rows/cols that are never stored.
  const int mA = min(m0 + r16, M - 1);
  const int c0 = n0 + r16;
  const int c1 = c0 + 16;
  const int c0c = min(c0, N - 1);
  const int c1c = min(c1, N - 1);
  v8f acc0 = {}, acc1 = {};
  const int k16 = K & ~15;
  const int kq  = K & ~3;
  int k0 = 0;
  for (; k0 < k16; k0 += 16) {
#pragma unroll
    for (int u = 0; u < 4; ++u) {
      const int ka = k0 + 4 * u + 2 * grp;
      v2f av = A.pair(b, mA, ka);
      v2f b0 = B.pair(b, ka, c0c);
      v2f b1 = B.pair(b, ka, c1c);
      acc0 = WMMA_F32(av, b0, acc0);
      acc1 = WMMA_F32(av, b1, acc1);
    }
  }
  for (; k0 < kq; k0 += 4) {
    const int ka = k0 + 2 * grp;
    v2f av = A.pair(b, mA, ka);
    v2f b0 = B.pair(b, ka, c0c);
    v2f b1 = B.pair(b, ka, c1c);
    acc0 = WMMA_F32(av, b0, acc0);
    acc1 = WMMA_F32(av, b1, acc1);
  }
  if (kq < K) {                        // ragged-K tail (K % 4 != 0), once
    const int ka = kq + 2 * grp;
    const int kb = ka + 1;
    v2f av = {0.f, 0.f}, b0 = {0.f, 0.f}, b1 = {0.f, 0.f};
    if (ka < K) {
      av.x = A.one(b, mA, ka); b0.x = B.one(b, ka, c0c); b1.x = B.one(b, ka, c1c);
    }
    if (kb < K) {
      av.y = A.one(b, mA, kb); b0.y = B.one(b, kb, c0c); b1.y = B.one(b, kb, c1c);
    }
    acc0 = WMMA_F32(av, b0, acc0);
    acc1 = WMMA_F32(av, b1, acc1);
  }
  // epilogue
  if (m0 + 15 < M) {                   // full-height tile: no row guards
    if (c0 < N) {
#pragma unroll
      for (int r = 0; r < 8; ++r) C(b, m0 + r + 8 * grp, c0, acc0[r]);
    }
    if (c1 < N) {
#pragma unroll
      for (int r = 0; r < 8; ++r) C(b, m0 + r + 8 * grp, c1, acc1[r]);
    }
  } else {
    if (c0 < N) {
#pragma unroll
      for (int r = 0; r < 8; ++r) {
        int m = m0 + r + 8 * grp;
        if (m < M) C(b, m, c0, acc0[r]);
      }
    }
    if (c1 < N) {
#pragma unroll
      for (int r = 0; r < 8; ++r) {
        int m = m0 + r + 8 * grp;
        if (m < M) C(b, m, c1, acc1[r]);
      }
    }
  }
}

// ---- softmax stats over the last axis, one block per row -------------------
__global__ __launch_bounds__(256) void row_stats_kernel(const float* __restrict__ p,
                                                        int n, float* mx,
                                                        float* inv) {
  __shared__ float red[256];
  const float* row = p + (long)blockIdx.x * n;
  float m = NEG_INF;
  for (int i = threadIdx.x; i < n; i += 256) m = fmaxf(m, row[i]);
  red[threadIdx.x] = m; __syncthreads();
  for (int s = 128; s; s >>= 1) {
    if (threadIdx.x < s) red[threadIdx.x] = fmaxf(red[threadIdx.x], red[threadIdx.x + s]);
    __syncthreads();
  }
  m = red[0]; __syncthreads();
  float sum = 0.f;
  for (int i = threadIdx.x; i < n; i += 256) sum += expf(row[i] - m);
  red[threadIdx.x] = sum; __syncthreads();
  for (int s = 128; s; s >>= 1) {
    if (threadIdx.x < s) red[threadIdx.x] += red[threadIdx.x + s];
    __syncthreads();
  }
  if (threadIdx.x == 0) { mx[blockIdx.x] = m; inv[blockIdx.x] = 1.0f / red[0]; }
}

// ---- per-pixel softmax stats over the class axis (stride HW) ---------------
__global__ void pix_stats_kernel(const float* __restrict__ score, int bs, int S,
                                 int HW, float* pmax, float* pinv) {
  long t = (long)blockIdx.x * blockDim.x + threadIdx.x;
  if (t >= (long)bs * HW) return;
  int b = (int)(t / HW), n = (int)(t % HW);
  const float* p = score + (long)b * S * HW + n;
  float m = NEG_INF;
  for (int s = 0; s < S; ++s) m = fmaxf(m, p[(long)s * HW]);
  float sum = 0.f;
  for (int s = 0; s < S; ++s) sum += expf(p[(long)s * HW] - m);
  pmax[t] = m; pinv[t] = 1.0f / sum;
}

// ---- GAT attention logits: s1[i]=Wh[i].a[:F], s2[i]=Wh[i].a[F:] ------------
__global__ void gat_scores_kernel(const float* __restrict__ Wh,
                                  const float* __restrict__ a, int G, int NG,
                                  int F, int amod, float* s1, float* s2) {
  int t = blockIdx.x * blockDim.x + threadIdx.x;
  if (t >= G * NG) return;
  int g = t / NG;
  const float* row = Wh + (long)t * F;
  const float* ap = a + (long)(g % amod) * (2 * F);
  float x1 = 0.f, x2 = 0.f;
  for (int f = 0; f < F; ++f) { float w = row[f]; x1 += w * ap[f]; x2 += w * ap[F + f]; }
  s1[t] = x1; s2[t] = x2;
}

// ---- att[i][j] = softmax_j(leaky_relu(s1[i]+s2[j])) ------------------------
__global__ __launch_bounds__(256) void gat_att_kernel(const float* __restrict__ s1,
                                                      const float* __restrict__ s2,
                                                      int NG, float* att) {
  __shared__ float vals[576];
  __shared__ float red[256];
  int rowg = blockIdx.x;            // g*NG + i
  int g = rowg / NG;
  float s1v = s1[rowg];
  const float* s2p = s2 + (long)g * NG;
  float m = NEG_INF;
  for (int j = threadIdx.x; j < NG; j += 256) {
    float v = s1v + s2p[j];
    v = (v >= 0.f) ? v : LRELU_ALPHA * v;
    vals[j] = v; m = fmaxf(m, v);
  }
  red[threadIdx.x] = m; __syncthreads();
  for (int s = 128; s; s >>= 1) {
    if (threadIdx.x < s) red[threadIdx.x] = fmaxf(red[threadIdx.x], red[threadIdx.x + s]);
    __syncthreads();
  }
  m = red[0]; __syncthreads();
  float sum = 0.f;
  for (int j = threadIdx.x; j < NG; j += 256) sum += expf(vals[j] - m);
  red[threadIdx.x] = sum; __syncthreads();
  for (int s = 128; s; s >>= 1) {
    if (threadIdx.x < s) red[threadIdx.x] += red[threadIdx.x + s];
    __syncthreads();
  }
  float inv = 1.0f / red[0];
  float* out = att + (long)rowg * NG;
  for (int j = threadIdx.x; j < NG; j += 256) out[j] = expf(vals[j] - m) * inv;
}

// ---- nodes = concat(region_nodes, class_nodes) along the node axis ---------
__global__ void build_nodes_kernel(const float* __restrict__ rn,
                                   const float* __restrict__ cn, int bs, int R,
                                   int S, int C, float* nodes) {
  long t = (long)blockIdx.x * blockDim.x + threadIdx.x;
  long total = (long)bs * (R + S) * C;
  if (t >= total) return;
  int c = (int)(t % C); long rr = t / C;
  int node = (int)(rr % (R + S)); int b = (int)(rr / (R + S));
  nodes[t] = (node < R) ? rn[((long)(b * R + node)) * C + c]
                        : cn[((long)(b * S + node - R)) * C + c];
}

// ---- x[b][i][h*Fp+f] = headout[b][h][i][f]  (elu applied in GEMM epilogue) -
__global__ void build_x_kernel(const float* __restrict__ ho, int bs, int H,
                               int NG, int Fp, float* x) {
  long t = (long)blockIdx.x * blockDim.x + threadIdx.x;
  long total = (long)bs * NG * H * Fp;
  if (t >= total) return;
  int f = (int)(t % Fp); long r = t / Fp;
  int hh = (int)(r % H); r /= H;
  int i = (int)(r % NG); int b = (int)(r / NG);
  x[t] = ho[(((long)(b * H + hh) * NG) + i) * Fp + f];
}

// ---- copy region_f into the first 1024 columns of the 1280-wide output -----
__global__ void copy_region_kernel(const float* __restrict__ rf, float* out,
                                   long rows, int incols, int outld) {
  long t = (long)blockIdx.x * blockDim.x + threadIdx.x;
  if (t >= rows * incols) return;
  long r = t / incols; int c = (int)(t % incols);
  out[r * outld + c] = rf[t];
}

// ---- 4x bilinear upsample (half-pixel centers, edge clamp == jax linear) ---
__global__ void resize_kernel(const float* __restrict__ seg, float* out, int bs,
                              int S, int h, int w, int oh, int ow) {
  long t = (long)blockIdx.x * blockDim.x + threadIdx.x;
  long total = (long)bs * S * oh * ow;
  if (t >= total) return;
  int ox = (int)(t % ow); long r = t / ow;
  int oy = (int)(r % oh); r /= oh;
  int o = (int)(r % S); int b = (int)(r / S);
  float iy = (oy + 0.5f) * ((float)h / (float)oh) - 0.5f;
  float ix = (ox + 0.5f) * ((float)w / (float)ow) - 0.5f;
  int y0 = (int)floorf(iy); float dy = iy - (float)y0;
  int x0 = (int)floorf(ix); float dx = ix - (float)x0;
  int y0c = min(max(y0, 0), h - 1), y1c = min(max(y0 + 1, 0), h - 1);
  int x0c = min(max(x0, 0), w - 1), x1c = min(max(x0 + 1, 0), w - 1);
  const float* sp = seg + (long)b * h * w * S;   // layout (b, y*w+x, o)
  float v00 = sp[((long)(y0c * w + x0c)) * S + o];
  float v01 = sp[((long)(y0c * w + x1c)) * S + o];
  float v10 = sp[((long)(y1c * w + x0c)) * S + o];
  float v11 = sp[((long)(y1c * w + x1c)) * S + o];
  out[t] = (1.f - dy) * ((1.f - dx) * v00 + dx * v01) +
           dy * ((1.f - dx) * v10 + dx * v11);
}

// ---------------------------------------------------------------------------
extern "C" void kernel_launch(void* const* d_in, const int* in_sizes, int n_in,
                              void* d_out, int out_size, void* d_ws,
                              size_t ws_size, hipStream_t stream) {
  (void)in_sizes; (void)n_in; (void)out_size; (void)ws_size;
  const float* region_f = (const float*)d_in[0];
  const float* stuff_f  = (const float*)d_in[1];
  const float* semseg   = (const float*)d_in[2];
  const float* W_rin  = (const float*)d_in[3];
  const float* b_rin  = (const float*)d_in[4];
  const float* W_sin  = (const float*)d_in[5];
  const float* b_sin  = (const float*)d_in[6];
  const float* W_heads = (const float*)d_in[7];
  const float* a_heads = (const float*)d_in[8];
  const float* W_gout = (const float*)d_in[9];
  const float* a_gout = (const float*)d_in[10];
  const float* W_rout = (const float*)d_in[11];
  const float* b_rout = (const float*)d_in[12];
  const float* W_sout = (const float*)d_in[13];
  const float* b_sout = (const float*)d_in[14];
  const float* W_seg  = (const float*)d_in[15];
  const float* b_seg  = (const float*)d_in[16];

  const int bs = 4, R = 512, S = 54, h = 100, w = 152, HW = h * w;
  const int fc = 1024, Cd = 512, Cg = 256, H = 4, Fp = 64, So = 128;
  const int NG = R + S;                       // 566 graph nodes
  const int G = bs * H;                       // 16 (batch, head) pairs
  const int oh = 4 * h, ow = 4 * w;

  // workspace layout (floats); total ~19.4M floats (~78 MB); all offsets even
  float* W = (float*)d_ws;
  long o = 0;
  float* rn   = W + o; o += (long)bs * R * Cg;        // region_nodes 2048x256
  float* cc   = W + o; o += (long)bs * S * Cd;        // class_center 216x512
  float* cn   = W + o; o += (long)bs * S * Cg;        // class_nodes  216x256
  float* rmx  = W + o; o += (long)bs * S;             // class-softmax max
  float* rinv = W + o; o += (long)bs * S + 2;         // class-softmax 1/sum (+pad)
  float* nodes= W + o; o += (long)bs * NG * Cg;       // nodes (later gnodes)
  float* Wh   = W + o; o += (long)G * NG * Fp;        // Wh (later Wh2: 4*566*256)
  float* s1   = W + o; o += (long)G * NG;
  float* s2   = W + o; o += (long)G * NG;
  float* att  = W + o; o += (long)G * NG * NG;        // att (later att2)
  float* ho   = W + o; o += (long)G * NG * Fp;        // head outputs (elu'ed)
  float* x    = W + o; o += (long)bs * NG * Cg;       // concat heads
  float* ncn  = W + o; o += (long)bs * S * So;        // new_class_nodes
  float* pmax = W + o; o += (long)bs * HW;
  float* pinv = W + o; o += (long)bs * HW;
  float* ns   = W + o; o += (long)bs * HW * So;       // new_stuff (b,n,o)
  float* seg  = W + o; o += (long)bs * HW * S;        // seg logits (b,n,o)
  float* Wh2 = Wh;        // alias (4*566*256 == 16*566*64)
  float* att2 = att;      // alias
  float* gnodes = nodes;  // alias

  float* out1 = (float*)d_out;                        // (2048, 1280)
  float* out2 = out1 + (long)bs * R * (fc + Cg);      // (4, 54, 400, 608)

  const BIdx one{1, BIGMOD, 0};

  // 1. region_nodes = region_f @ W_rin + b_rin      [2048 x 1024 x 256]
  {
    PlainA A{region_f, fc, one};
    PlainB B{W_rin, Cg, one};
    StoreC<0> Cst{rn, Cg, one, b_rin};
    wmma_gemm<<<dim3(cdiv(Cg, 32), cdiv(bs * R, 128), 1), 256, 0, stream>>>(
        A, B, Cst, bs * R, Cg, fc);
  }
  // 2. per-(b,s) softmax stats over n
  row_stats_kernel<<<bs * S, 256, 0, stream>>>(semseg, HW, rmx, rinv);
  // 3. class_center[b,s,c] = sum_n softmax(score) * stuff   [54 x 15200 x 512]
  {
    ExpRowA A{semseg, HW, (long)S * HW, rmx, rinv, S};
    TransB B{stuff_f, HW, {1, BIGMOD, (long)Cd * HW}};
    StoreC<0> Cst{cc, Cd, {1, BIGMOD, (long)S * Cd}, nullptr};
    wmma_gemm<<<dim3(cdiv(Cd, 32), cdiv(S, 128), bs), 256, 0, stream>>>(
        A, B, Cst, S, Cd, HW);
  }
  // 4. class_nodes = class_center @ W_sin + b_sin   [216 x 512 x 256]
  {
    PlainA A{cc, Cd, one};
    PlainB B{W_sin, Cg, one};
    StoreC<0> Cst{cn, Cg, one, b_sin};
    wmma_gemm<<<dim3(cdiv(Cg, 32), cdiv(bs * S, 128), 1), 256, 0, stream>>>(
        A, B, Cst, bs * S, Cg, Cd);
  }
  // 5. nodes = concat(region_nodes, class_nodes)
  {
    long total = (long)bs * NG * Cg;
    build_nodes_kernel<<<(int)((total + 255) / 256), 256, 0, stream>>>(
        rn, cn, bs, R, S, Cg, nodes);
  }
  // 6. Wh[b,h] = nodes[b] @ W_heads[h]              [566 x 256 x 64, z=16]
  {
    PlainA A{nodes, Cg, {H, BIGMOD, (long)NG * Cg}};
    PlainB B{W_heads, Fp, {1, H, (long)Cg * Fp}};
    StoreC<0> Cst{Wh, Fp, {1, BIGMOD, (long)NG * Fp}, nullptr};
    wmma_gemm<<<dim3(cdiv(Fp, 32), cdiv(NG, 128), G), 256, 0, stream>>>(
        A, B, Cst, NG, Fp, Cg);
  }
  // 7-8. attention logits + row softmax (heads)
  gat_scores_kernel<<<cdiv(G * NG, 256), 256, 0, stream>>>(Wh, a_heads, G, NG,
                                                           Fp, H, s1, s2);
  gat_att_kernel<<<G * NG, 256, 0, stream>>>(s1, s2, NG, att);
  // 9. headout = elu(att @ Wh)                      [566 x 566 x 64, z=16]
  {
    PlainA A{att, NG, {1, BIGMOD, (long)NG * NG}};
    PlainB B{Wh, Fp, {1, BIGMOD, (long)NG * Fp}};
    StoreC<1> Cst{ho, Fp, {1, BIGMOD, (long)NG * Fp}, nullptr};
    wmma_gemm<<<dim3(cdiv(Fp, 32), cdiv(NG, 128), G), 256, 0, stream>>>(
        A, B, Cst, NG, Fp, NG);
  }
  // 10. x = interleave heads
  {
    long total = (long)bs * NG * Cg;
    build_x_kernel<<<(int)((total + 255) / 256), 256, 0, stream>>>(ho, bs, H,
                                                                   NG, Fp, x);
  }
  // 11. Wh2 = x @ W_gout                            [566 x 256 x 256, z=4]
  {
    PlainA A{x, Cg, {1, BIGMOD, (long)NG * Cg}};
    PlainB B{W_gout, Cg, one};
    StoreC<0> Cst{Wh2, Cg, {1, BIGMOD, (long)NG * Cg}, nullptr};
    wmma_gemm<<<dim3(cdiv(Cg, 32), cdiv(NG, 128), bs), 256, 0, stream>>>(
        A, B, Cst, NG, Cg, Cg);
  }
  // 12-13. output-layer attention
  gat_scores_kernel<<<cdiv(bs * NG, 256), 256, 0, stream>>>(Wh2, a_gout, bs,
                                                            NG, Cg, 1, s1, s2);
  gat_att_kernel<<<bs * NG, 256, 0, stream>>>(s1, s2, NG, att2);
  // 14. gnodes = elu(att2 @ Wh2)                    [566 x 566 x 256, z=4]
  {
    PlainA A{att2, NG, {1, BIGMOD, (long)NG * NG}};
    PlainB B{Wh2, Cg, {1, BIGMOD, (long)NG * Cg}};
    StoreC<1> Cst{gnodes, Cg, {1, BIGMOD, (long)NG * Cg}, nullptr};
    wmma_gemm<<<dim3(cdiv(Cg, 32), cdiv(NG, 128), bs), 256, 0, stream>>>(
        A, B, Cst, NG, Cg, NG);
  }
  // 15. out1[:, :1024] = region_f
  {
    long total = (long)bs * R * fc;
    copy_region_kernel<<<(int)((total + 255) / 256), 256, 0, stream>>>(
        region_f, out1, (long)bs * R, fc, fc + Cg);
  }
  // 16. out1[:, 1024:] = new_region @ W_rout + b_rout   [512 x 256 x 256, z=4]
  {
    PlainA A{gnodes, Cg, {1, BIGMOD, (long)NG * Cg}};   // rows 0..511 per batch
    PlainB B{W_rout, Cg, one};
    StoreC<0> Cst{out1 + fc, fc + Cg, {1, BIGMOD, (long)R * (fc + Cg)}, b_rout};
    wmma_gemm<<<dim3(cdiv(Cg, 32), cdiv(R, 128), bs), 256, 0, stream>>>(
        A, B, Cst, R, Cg, Cg);
  }
  // 17. ncn = new_class @ W_sout + b_sout           [54 x 256 x 128, z=4]
  {
    PlainA A{gnodes + (long)R * Cg, Cg, {1, BIGMOD, (long)NG * Cg}};
    PlainB B{W_sout, So, one};
    StoreC<0> Cst{ncn, So, {1, BIGMOD, (long)S * So}, b_sout};
    wmma_gemm<<<dim3(cdiv(So, 32), cdiv(S, 128), bs), 256, 0, stream>>>(
        A, B, Cst, S, So, Cg);
  }
  // 18. per-pixel softmax stats over classes
  pix_stats_kernel<<<cdiv(bs * HW, 256), 256, 0, stream>>>(semseg, bs, S, HW,
                                                           pmax, pinv);
  // 19. new_stuff[b,n,o] = sum_s softmax_s(score) * ncn  [15200 x 54 x 128, z=4]
  {
    ExpColA A{semseg, HW, (long)S * HW, pmax, pinv, HW};
    PlainB B{ncn, So, {1, BIGMOD, (long)S * So}};
    StoreC<0> Cst{ns, So, {1, BIGMOD, (long)HW * So}, nullptr};
    wmma_gemm<<<dim3(cdiv(So, 32), cdiv(HW, 128), bs), 256, 0, stream>>>(
        A, B, Cst, HW, So, S);
  }
  // 20. seg[b,n,o] = concat(stuff,new_stuff)[n,:] @ W_seg + b_seg
  //                                                  [15200 x 640 x 54, z=4]
  {
    SegA A{stuff_f, ns, HW, Cd, So};
    PlainB B{W_seg, S, one};
    StoreC<0> Cst{seg, S, {1, BIGMOD, (long)HW * S}, b_seg};
    wmma_gemm<<<dim3(cdiv(S, 32), cdiv(HW, 128), bs), 256, 0, stream>>>(
        A, B, Cst, HW, S, Cd + So);
  }
  // 21. 4x bilinear upsample -> out2 (4,54,400,608)
  {
    long total = (long)bs * S * oh * ow;
    resize_kernel<<<(int)((total + 255) / 256), 256, 0, stream>>>(
        seg, out2, bs, S, h, w, oh, ow);
  }
}